// A_55439437856806
// MI455X (gfx1250) — compile-verified
//
#include <hip/hip_runtime.h>

#define N_VOXD 256
#define N_ANGD 128
#define ROWF   260          // LDS row stride in floats = 256 + 4 pad (TDM pad_amount)
#define HALFV  127.5f
#define NTHR   512          // 16 waves -> 4 waves per SIMD32, 1 WG per WGP (LDS-bound)

typedef float v2f __attribute__((ext_vector_type(2)));
typedef float v8f __attribute__((ext_vector_type(8)));
typedef unsigned int u32x4 __attribute__((ext_vector_type(4)));
typedef int i32x4 __attribute__((ext_vector_type(4)));
typedef int i32x8 __attribute__((ext_vector_type(8)));

#if defined(__gfx1250__) && __has_builtin(__builtin_amdgcn_wmma_f32_16x16x4_f32)
#define USE_WMMA 1
#else
#define USE_WMMA 0
#endif

#if defined(__gfx1250__) && __has_builtin(__builtin_amdgcn_tensor_load_to_lds) && \
    __has_builtin(__builtin_amdgcn_s_wait_tensorcnt)
#define USE_TDM 1
#else
#define USE_TDM 0
#endif

__global__ __launch_bounds__(NTHR) void radon_slice_kernel(
    const float* __restrict__ vol, const float* __restrict__ angles,
    float* __restrict__ out) {
  __shared__ float smem[N_VOXD * ROWF];   // 266,240 B of the 320 KB WGP LDS

  const int z    = blockIdx.x;
  const int a    = blockIdx.y;
  const int tid  = threadIdx.x;
  const int lane = tid & 31;
  const int wave = tid >> 5;
  const int n    = lane & 15;          // N (=u) index within 16x16 tile
  const bool lo  = (lane < 16);

  const float* g = vol + (size_t)z * (N_VOXD * N_VOXD);

#if USE_TDM
  // ---- Phase 1 (TDM): one tensor DMA stages the whole 256x256 f32 slice into
  // LDS; pad_interval=7 (every 256 DWORDs) / pad_amount=3 (4 DWORDs) lays it out
  // with the bank-conflict-free 260-float row stride. Passing smem's real LDS
  // address into the descriptor also makes the LDS object "escape" so the
  // compiler cannot fold the (hardware-written) LDS loads to undef. ----
  const unsigned lds_base = (unsigned)(unsigned long long)(const void*)smem;
  if (wave == 0) {
    const unsigned long long ga = (unsigned long long)g;
    u32x4 g0;
    g0[0] = 1u;                                   // count=1, user descriptor
    g0[1] = lds_base;                             // lds_addr (byte offset of smem)
    g0[2] = (unsigned int)(ga & 0xFFFFFFFFull);   // global_addr[31:0]
    g0[3] = (unsigned int)((ga >> 32) & 0x1FFFFFFull) | (2u << 30); // addr hi | type=2

    i32x8 g1;
    g1[0] = (2 << 16)      // data_size = 4B
          | (1 << 20)      // pad_enable
          | (7 << 22)      // pad_interval: every 256 DWORDs (one row)
          | (3 << 25);     // pad_amount: 4 DWORDs -> 260-float row stride
    g1[1] = 256 << 16;     // tensor_dim0 = 256 (low 16 bits in [63:48])
    g1[2] = 256 << 16;     // tensor_dim0 hi=0 | tensor_dim1 = 256 (low 16)
    g1[3] = 256 << 16;     // tensor_dim1 hi=0 | tile_dim0 = 256
    g1[4] = 256;           // tile_dim1 = 256, tile_dim2 = 0
    g1[5] = 256;           // tensor_dim0_stride = 256 (low 32)
    g1[6] = 0;             // stride0 hi | tensor_dim1_stride low 16
    g1[7] = 1;             // tensor_dim1_stride = 65536 (bit 16 of its 48b field)

    i32x4 g2 = {0, 0, 0, 0};
    i32x4 g3 = {0, 0, 0, 0};
#if __clang_major__ >= 23
    i32x8 g4 = {0, 0, 0, 0, 0, 0, 0, 0};
    __builtin_amdgcn_tensor_load_to_lds(g0, g1, g2, g3, g4, 0);
#else
    __builtin_amdgcn_tensor_load_to_lds(g0, g1, g2, g3, 0);
#endif
    __builtin_amdgcn_s_wait_tensorcnt(0);
    asm volatile("" ::: "memory");   // DMA wrote LDS: fence the compiler's view
  }
#else
  // ---- Phase 1 (fallback): coalesced float4 staging ----
  #pragma unroll 4
  for (int k = 0; k < 32; ++k) {
    const int idx  = k * (NTHR * 4) + tid * 4;
    const int y    = idx >> 8;
    const int xcol = idx & 255;            // multiple of 4 -> 16B-aligned LDS store
    const float4 v = *(const float4*)(g + idx);
    *(float4*)(&smem[y * ROWF + xcol]) = v;
  }
#endif
  __syncthreads();

  const float th = angles[a];
  const float ct = cosf(th), st = sinf(th);

  // ---- Phase 2: wave w owns u-tile w; loop all 16 t-tiles ----
  const int ut = wave;
  const float uval = (float)(ut * 16 + n) - HALFV;
  float acc = 0.f;

#if USE_WMMA
  // B (4x16): row0 = ct/st, row1 = affine offset, rows 2,3 = 0 (upper lanes)
  v2f bx, by;
  bx.x = lo ? ct : 0.f;  bx.y = lo ? (HALFV - st * uval) : 0.f;
  by.x = lo ? st : 0.f;  by.y = lo ? (HALFV + ct * uval) : 0.f;
#endif

  for (int tt = 0; tt < 16; ++tt) {
    v8f xsv, ysv;
#if USE_WMMA
    // A (16x4): row m = [t_m, 1, 0, 0]; lanes 0-15 hold K=0,1; 16-31 hold K=2,3(=0)
    v2f av;
    av.x = lo ? ((float)(tt * 16 + lane) - HALFV) : 0.f;
    av.y = lo ? 1.f : 0.f;
    v8f cz = {};
    // xs(t,u) = t*ct + (half - st*u); ys(t,u) = t*st + (half + ct*u)
    xsv = __builtin_amdgcn_wmma_f32_16x16x4_f32(false, av, false, bx,
                                                (short)0, cz, false, false);
    ysv = __builtin_amdgcn_wmma_f32_16x16x4_f32(false, av, false, by,
                                                (short)0, cz, false, false);
#else
    #pragma unroll
    for (int j = 0; j < 8; ++j) {
      const float tval = (float)(tt * 16 + j + (lo ? 0 : 8)) - HALFV;
      xsv[j] = tval * ct - uval * st + HALFV;
      ysv[j] = tval * st + uval * ct + HALFV;
    }
#endif
    // Consume in C/D layout: lane -> N=u (=lane%16), VGPR j -> M=t row (+8 upper half)
    #pragma unroll
    for (int j = 0; j < 8; ++j) {
      const float xs = xsv[j], ys = ysv[j];
      const float x0f = floorf(xs), y0f = floorf(ys);
      const float fx = xs - x0f, fy = ys - y0f;
      const int x0 = (int)x0f, y0 = (int)y0f;
      const int x1 = x0 + 1,  y1 = y0 + 1;
      const float mx0 = ((unsigned)x0 < 256u) ? 1.f : 0.f;
      const float mx1 = ((unsigned)x1 < 256u) ? 1.f : 0.f;
      const float my0 = ((unsigned)y0 < 256u) ? 1.f : 0.f;
      const float my1 = ((unsigned)y1 < 256u) ? 1.f : 0.f;
      const int xc0 = min(max(x0, 0), 255), xc1 = min(max(x1, 0), 255);
      const int yc0 = min(max(y0, 0), 255), yc1 = min(max(y1, 0), 255);
      const float v00 = smem[yc0 * ROWF + xc0];
      const float v01 = smem[yc0 * ROWF + xc1];
      const float v10 = smem[yc1 * ROWF + xc0];
      const float v11 = smem[yc1 * ROWF + xc1];
      const float gx = 1.f - fx, gy = 1.f - fy;
      acc += v00 * (gx * gy * mx0 * my0)
           + v01 * (fx * gy * mx1 * my0)
           + v10 * (gx * fy * mx0 * my1)
           + v11 * (fx * fy * mx1 * my1);
    }
  }

  // Fold M=0..7 (lanes 0-15) with M=8..15 (lanes 16-31): same u, xor-16 partner
  acc += __shfl_xor(acc, 16, 32);
  if (lo) {
    out[((size_t)a * N_VOXD + z) * N_VOXD + (ut * 16 + n)] = acc;
  }
}

extern "C" void kernel_launch(void* const* d_in, const int* in_sizes, int n_in,
                              void* d_out, int out_size, void* d_ws, size_t ws_size,
                              hipStream_t stream) {
  (void)in_sizes; (void)n_in; (void)d_ws; (void)ws_size; (void)out_size;
  const float* x      = (const float*)d_in[0];
  const float* angles = (const float*)d_in[1];
  float* out          = (float*)d_out;

  dim3 grid(N_VOXD, N_ANGD);   // (z, angle)
  radon_slice_kernel<<<grid, NTHR, 0, stream>>>(x, angles, out);
}